// DreamCore_7464653160652
// MI455X (gfx1250) — compile-verified
//
#include <hip/hip_runtime.h>
#include <math.h>

// ---- problem constants (match reference) ----
#define D_MODEL 8192
#define D_HALF  4096
#define N_FRAG  64
#define STEPS   20
#define NTILES  (D_HALF / 16)   // 256 column tiles of 16

typedef __attribute__((ext_vector_type(2))) float v2f;
typedef __attribute__((ext_vector_type(8))) float v8f;

__device__ __forceinline__ float gelu_tanh(float x) {
    // jax.nn.gelu default (approximate=True)
    const float c = 0.7978845608028654f; // sqrt(2/pi)
    float x3 = x * x * x;
    float t = tanhf(c * (x + 0.044715f * x3));
    return 0.5f * x * (1.0f + t);
}

// ---------------------------------------------------------------------------
// Kernel A: fused (F @ w1 + b1) -> gelu -> *w2 -> partial row sums per N-tile
// grid = 128 blocks, block = 256 threads (8 waves, wave32).
// Each block owns a 32-column (128-byte) slab of w1 so every L2 line of the
// streamed 134 MB w1 is fully consumed within one WGP:
//   waves 0-3: columns [32b,    32b+16), M tiles 0..3
//   waves 4-7: columns [32b+16, 32b+32), M tiles 0..3
// Inner product via V_WMMA_F32_16X16X4_F32 over K=8192, two acc chains.
// ---------------------------------------------------------------------------
__global__ __launch_bounds__(256)
void fused_logits_partial(const float* __restrict__ F,      // (64, 8192)
                          const float* __restrict__ w1,     // (8192, 4096)
                          const float* __restrict__ bias1,  // (4096,)
                          const float* __restrict__ wvec2,  // (4096,)
                          float* __restrict__ partial)      // (256, 64)
{
    const int lane  = threadIdx.x & 31;
    const int wave  = threadIdx.x >> 5;         // 0..7
    const int half  = lane >> 4;                // 0: lanes 0-15, 1: lanes 16-31
    const int l     = lane & 15;
    const int mtile = wave & 3;
    const int m0    = mtile * 16;
    const int ntile = blockIdx.x * 2 + (wave >> 2);
    const int n0    = ntile * 16;
    const int col   = n0 + l;

    // A: row (m0+l) of fragments; this lane supplies K = {k+2*half, k+1+2*half}
    const float* __restrict__ Arow = F + (size_t)(m0 + l) * D_MODEL + 2 * half;
    // B: column `col` of w1; this lane supplies rows K = {k+2*half, k+1+2*half}
    const float* __restrict__ Bcol = w1 + (size_t)(2 * half) * D_HALF + col;

    v8f acc0 = {};
    v8f acc1 = {};

    for (int k = 0; k < D_MODEL; k += 8) {
        // prefetch the strided w1 rows a couple of steps ahead
        __builtin_prefetch(Bcol + (size_t)(k + 16) * D_HALF, 0, 1);

        v2f a0 = *(const v2f*)(Arow + k);
        v2f a1 = *(const v2f*)(Arow + k + 4);

        v2f b0, b1;
        b0.x = Bcol[(size_t)(k + 0) * D_HALF];
        b0.y = Bcol[(size_t)(k + 1) * D_HALF];
        b1.x = Bcol[(size_t)(k + 4) * D_HALF];
        b1.y = Bcol[(size_t)(k + 5) * D_HALF];

        // D = A x B + C   (f32, 16x16x4), two independent chains for ILP
        acc0 = __builtin_amdgcn_wmma_f32_16x16x4_f32(
                   false, a0, false, b0, (short)0, acc0, false, false);
        acc1 = __builtin_amdgcn_wmma_f32_16x16x4_f32(
                   false, a1, false, b1, (short)0, acc1, false, false);
    }

    const float bv = bias1[col];
    const float wv = wvec2[col];

    // C/D layout: VGPR r holds (M = m0 + r + 8*half, N = col) on this lane.
    #pragma unroll
    for (int r = 0; r < 8; ++r) {
        float v = acc0[r] + acc1[r] + bv;
        float p = gelu_tanh(v) * wv;
        // reduce across the 16 lanes of this half (xor masks stay in-half)
        p += __shfl_xor(p, 1);
        p += __shfl_xor(p, 2);
        p += __shfl_xor(p, 4);
        p += __shfl_xor(p, 8);
        if (l == 0) {
            partial[(size_t)ntile * N_FRAG + (m0 + r + 8 * half)] = p;
        }
    }
}

// ---------------------------------------------------------------------------
// Kernel B: deterministic partial reduction + bias + softmax over 64 logits
// grid = 1, block = 64
// ---------------------------------------------------------------------------
__global__ void softmax_kernel(const float* __restrict__ partial, // (256,64)
                               const float* __restrict__ b2,      // (1,)
                               float* __restrict__ weights_out,   // d_out slice
                               float* __restrict__ weights_ws,    // ws copy
                               float* __restrict__ wsum_ws)       // ws scalar
{
    __shared__ float s[N_FRAG];
    const int t = threadIdx.x;

    float lg = b2[0];
    for (int nt = 0; nt < NTILES; ++nt)
        lg += partial[(size_t)nt * N_FRAG + t];

    s[t] = lg;
    __syncthreads();
    float mx = -INFINITY;
    for (int i = 0; i < N_FRAG; ++i) mx = fmaxf(mx, s[i]);
    float e = __expf(lg - mx);
    __syncthreads();
    s[t] = e;
    __syncthreads();
    float denom = 0.0f;
    for (int i = 0; i < N_FRAG; ++i) denom += s[i];

    float w = e / denom;
    weights_out[t] = w;
    weights_ws[t]  = w;

    __syncthreads();
    s[t] = w;
    __syncthreads();
    if (t == 0) {
        float ws = 0.0f;
        for (int i = 0; i < N_FRAG; ++i) ws += s[i];
        *wsum_ws = ws;   // == sum(weights), matches reference numerically
    }
}

// ---------------------------------------------------------------------------
// Kernel C: wf[d] = sum_i weights[i] * F[i][d]
// grid = 32, block = 256
// ---------------------------------------------------------------------------
__global__ __launch_bounds__(256)
void wf_kernel(const float* __restrict__ F,
               const float* __restrict__ weights,
               float* __restrict__ wf)
{
    const int d = blockIdx.x * blockDim.x + threadIdx.x;
    float acc = 0.0f;
    #pragma unroll 4
    for (int i = 0; i < N_FRAG; ++i)
        acc += weights[i] * F[(size_t)i * D_MODEL + d];
    wf[d] = acc;
}

// ---------------------------------------------------------------------------
// Kernel D: 20 diffusion steps, independent per dimension. Writes trajectory
// rows 0..20 and new_concept.   grid = 32, block = 256
// ---------------------------------------------------------------------------
__global__ __launch_bounds__(256)
void diffuse_kernel(const float* __restrict__ x0,
                    const float* __restrict__ noise,   // (20, 8192)
                    const float* __restrict__ wf,
                    const float* __restrict__ wsum_p,
                    float* __restrict__ new_concept,   // (8192)
                    float* __restrict__ trajectory)    // (21, 8192)
{
    const int d = blockIdx.x * blockDim.x + threadIdx.x;
    const float dt    = 1.0f / (float)STEPS;
    const float sigma = sqrtf(2.0f * 0.1f * dt);
    const float wsum  = *wsum_p;
    const float wfv   = wf[d];

    float x = x0[d];
    trajectory[d] = x;
    for (int t = 0; t < STEPS; ++t) {
        float g = 2.0f * (wsum * x - wfv) + 0.02f * x;
        x = x - dt * g + sigma * noise[(size_t)t * D_MODEL + d];
        trajectory[(size_t)(t + 1) * D_MODEL + d] = x;
    }
    new_concept[d] = x;
}

// ---------------------------------------------------------------------------
// Kernel E: per-fragment weighted squared distance. grid = 64, block = 256
// ---------------------------------------------------------------------------
__global__ __launch_bounds__(256)
void energy_partial_kernel(const float* __restrict__ x,
                           const float* __restrict__ F,
                           const float* __restrict__ weights,
                           float* __restrict__ epart)    // (64,)
{
    __shared__ float red[256];
    const int i = blockIdx.x;
    const int t = threadIdx.x;
    const float* __restrict__ Fi = F + (size_t)i * D_MODEL;

    float acc = 0.0f;
    for (int d = t; d < D_MODEL; d += 256) {
        float diff = x[d] - Fi[d];
        acc += diff * diff;
    }
    red[t] = acc;
    __syncthreads();
    for (int off = 128; off > 0; off >>= 1) {
        if (t < off) red[t] += red[t + off];
        __syncthreads();
    }
    if (t == 0) epart[i] = weights[i] * red[0];
}

// Kernel F: ordered final sum (deterministic). grid = 1, block = 1
__global__ void energy_final_kernel(const float* __restrict__ epart,
                                    float* __restrict__ energy_out)
{
    float s = 0.0f;
    for (int i = 0; i < N_FRAG; ++i) s += epart[i];
    *energy_out = s;
}

// ---------------------------------------------------------------------------
extern "C" void kernel_launch(void* const* d_in, const int* in_sizes, int n_in,
                              void* d_out, int out_size, void* d_ws, size_t ws_size,
                              hipStream_t stream) {
    const float* F     = (const float*)d_in[0];  // (64, 8192)
    const float* x0    = (const float*)d_in[1];  // (8192,)
    const float* noise = (const float*)d_in[2];  // (20, 8192)
    const float* w1    = (const float*)d_in[3];  // (8192, 4096)
    const float* b1    = (const float*)d_in[4];  // (4096,)
    const float* w2    = (const float*)d_in[5];  // (4096,)
    const float* b2    = (const float*)d_in[6];  // (1,)

    float* out = (float*)d_out;
    // return order: new_concept (8192), trajectory (21*8192), weights (64), energy (1)
    float* out_concept = out;
    float* out_traj    = out + D_MODEL;
    float* out_weights = out + D_MODEL + (STEPS + 1) * D_MODEL;
    float* out_energy  = out_weights + N_FRAG;

    // workspace layout (floats)
    float* ws          = (float*)d_ws;
    float* ws_partial  = ws;                         // 256*64
    float* ws_weights  = ws + NTILES * N_FRAG;       // 64
    float* ws_wsum     = ws_weights + N_FRAG;        // 1
    float* ws_wf       = ws_wsum + 63;               // 8192 (padded start)
    float* ws_epart    = ws_wf + D_MODEL;            // 64

    fused_logits_partial<<<NTILES / 2, 256, 0, stream>>>(F, w1, b1, w2, ws_partial);
    softmax_kernel<<<1, N_FRAG, 0, stream>>>(ws_partial, b2, out_weights,
                                             ws_weights, ws_wsum);
    wf_kernel<<<D_MODEL / 256, 256, 0, stream>>>(F, ws_weights, ws_wf);
    diffuse_kernel<<<D_MODEL / 256, 256, 0, stream>>>(x0, noise, ws_wf, ws_wsum,
                                                      out_concept, out_traj);
    energy_partial_kernel<<<N_FRAG, 256, 0, stream>>>(out_concept, F,
                                                      ws_weights, ws_epart);
    energy_final_kernel<<<1, 1, 0, stream>>>(ws_epart, out_energy);
}